// LinkPredictor_10651518894409
// MI455X (gfx1250) — compile-verified
//
#include <hip/hip_runtime.h>

// CDNA5 / gfx1250: wave32, V_WMMA_F32_16X16X4_F32 + TENSOR_LOAD_TO_LDS (TDM)
typedef __attribute__((ext_vector_type(2))) float v2f;   // WMMA A/B: 16x4 / 4x16 f32
typedef __attribute__((ext_vector_type(8))) float v8f;   // WMMA C/D: 16x16 f32
typedef __attribute__((ext_vector_type(4))) unsigned int u32x4;
typedef __attribute__((ext_vector_type(8))) int i32x8;
typedef __attribute__((ext_vector_type(4))) int i32x4;

#define HIDF 64

// --- TDM: DMA a [dim1 x 64] f32 row-major matrix from global into LDS (offset lds_off).
// D# per CDNA5 ISA ch.8: group0 = {count, lds_addr, global_addr|type=2},
// group1 = {data_size=4B, tensor_dim0=64, tensor_dim1=dim1, tile_dim0=64, tile_dim1=dim1,
//           tensor_dim0_stride=64}. Tile == tensor -> one full-matrix DMA. Tracked by TENSORcnt.
__device__ __forceinline__ void tdm_load_mat_to_lds(const float* gptr, unsigned lds_off, int dim1) {
  const unsigned long long ga = (unsigned long long)(const void*)gptr;
  u32x4 g0;
  g0[0] = 1u;                                              // count=1 (valid user descriptor)
  g0[1] = lds_off;                                         // lds_addr (bytes)
  g0[2] = (unsigned)(ga & 0xffffffffu);                    // global_addr[31:0]
  g0[3] = (unsigned)((ga >> 32) & 0x1ffffffu) | (2u << 30); // global_addr[56:32] | type=2
  i32x8 g1;
  g1[0] = (2 << 16);             // workgroup_mask=0, data_size=2 (4 bytes)
  g1[1] = (HIDF & 0xffff) << 16; // bits[63:48]  = tensor_dim0 = 64
  g1[2] = (dim1 & 0xffff) << 16; // bits[95:80]  = tensor_dim1 = dim1
  g1[3] = (HIDF & 0xffff) << 16; // bits[127:112]= tile_dim0   = 64
  g1[4] = (dim1 & 0xffff);       // bits[143:128]= tile_dim1   = dim1 (tile_dim2 = 0)
  g1[5] = HIDF;                  // bits[191:160]= tensor_dim0_stride = 64
  g1[6] = 0;
  g1[7] = 0;
  i32x4 z4 = {0, 0, 0, 0};
#if __clang_major__ >= 23
  i32x8 z8 = {0, 0, 0, 0, 0, 0, 0, 0};
  __builtin_amdgcn_tensor_load_to_lds(g0, g1, z4, z4, z8, 0);
#else
  __builtin_amdgcn_tensor_load_to_lds(g0, g1, z4, z4, 0);
#endif
}

__global__ void zero_f32(float* __restrict__ p, long n) {
  long i = (long)blockIdx.x * blockDim.x + threadIdx.x;
  if (i < n) p[i] = 0.0f;
}

__global__ void degree_kernel(const int* __restrict__ dst, float* __restrict__ deg, int E) {
  int e = blockIdx.x * blockDim.x + threadIdx.x;
  if (e < E) atomicAdd(&deg[dst[e]], 1.0f);
}

// Ya = X @ Wa ; Yb = X @ Wb + bias   (X: [rows,K], Wa/Wb: [K,64], row-major)
// Weights staged into LDS once per block via TDM; inner loop: A from global (streaming),
// B from LDS. One wave per 16-row tile; 4 col-tiles; f32 WMMA, K-step 4.
__global__ void gemm_dual_wmma(const float* __restrict__ X,
                               const float* __restrict__ Wa,
                               const float* __restrict__ Wb,
                               const float* __restrict__ bias,
                               float* __restrict__ Ya,
                               float* __restrict__ Yb,
                               int rows, int K) {
  extern __shared__ float smem[];           // [K*64] Wa | [K*64] Wb
  const int wave = threadIdx.x >> 5;
  const int lane = threadIdx.x & 31;
  const int half = lane >> 4;               // 0: K=k0,k0+1  1: K=k0+2,k0+3
  const int l16  = lane & 15;

  if (wave == 0) {
    tdm_load_mat_to_lds(Wa, 0u, K);
    tdm_load_mat_to_lds(Wb, (unsigned)(K * HIDF * 4), K);
    __builtin_amdgcn_s_wait_tensorcnt(0);
  }
  __syncthreads();                          // all waves participate BEFORE any tail-return

  const int rowBase = (blockIdx.x * (blockDim.x >> 5) + wave) * 16;
  if (rowBase >= rows) return;              // per-wave uniform: EXEC all-ones for WMMA

  const float* __restrict__ Wa_s = smem;
  const float* __restrict__ Wb_s = smem + (size_t)K * HIDF;
  const float* xrow = X + (size_t)(rowBase + l16) * K;

  v8f accA[4] = {};
  v8f accB[4] = {};
  for (int k0 = 0; k0 < K; k0 += 4) {
    const int ka = k0 + half * 2;
    v2f a;
    a.x = xrow[ka];
    a.y = xrow[ka + 1];
#pragma unroll
    for (int t = 0; t < 4; ++t) {
      const int col = t * 16 + l16;
      v2f bA, bB;
      bA.x = Wa_s[ka * HIDF + col];
      bA.y = Wa_s[(ka + 1) * HIDF + col];
      accA[t] = __builtin_amdgcn_wmma_f32_16x16x4_f32(false, a, false, bA, (short)0,
                                                      accA[t], false, false);
      bB.x = Wb_s[ka * HIDF + col];
      bB.y = Wb_s[(ka + 1) * HIDF + col];
      accB[t] = __builtin_amdgcn_wmma_f32_16x16x4_f32(false, a, false, bB, (short)0,
                                                      accB[t], false, false);
    }
  }
#pragma unroll
  for (int t = 0; t < 4; ++t) {
    const int col = t * 16 + l16;
    const float bb = bias[col];
#pragma unroll
    for (int r = 0; r < 8; ++r) {
      const size_t row = (size_t)rowBase + r + half * 8;  // C/D: VGPR r -> M=r / M=r+8
      Ya[row * HIDF + col] = accA[t][r];
      Yb[row * HIDF + col] = accB[t][r] + bb;
    }
  }
}

// Edge scatter-add of pre-transformed 64-dim features: 16 lanes/edge, float4/lane.
// Prefetch a look-ahead edge's source row (random L2 hits -> keep LOADcnt queue full).
__global__ void scatter_add64(const float* __restrict__ feat,
                              const int* __restrict__ src,
                              const int* __restrict__ dst,
                              float* __restrict__ agg, int E) {
  long t = (long)blockIdx.x * blockDim.x + threadIdx.x;
  int e = (int)(t >> 4);
  if (e >= E) return;
  int c = ((int)t & 15) * 4;
  int ep = e + 256;
  if (ep < E) __builtin_prefetch(feat + (size_t)src[ep] * HIDF + c, 0, 1);
  const float4 v = *(const float4*)(feat + (size_t)src[e] * HIDF + c);
  float* o = agg + (size_t)dst[e] * HIDF + c;
  atomicAdd(o + 0, v.x);
  atomicAdd(o + 1, v.y);
  atomicAdd(o + 2, v.z);
  atomicAdd(o + 3, v.w);
}

// out = [relu]( agg/max(deg,1) + selfpart )
template <bool RELU>
__global__ void finalize_kernel(const float* __restrict__ agg,
                                const float* __restrict__ selfp,
                                const float* __restrict__ deg,
                                float* __restrict__ out, long total) {
  long i = (long)blockIdx.x * blockDim.x + threadIdx.x;
  if (i >= total) return;
  const float dg = fmaxf(deg[i >> 6], 1.0f);
  float v = agg[i] / dg + selfp[i];
  if (RELU) v = fmaxf(v, 0.0f);
  out[i] = v;
}

// Decoder: wave computes 16 pairs. A = [z_s | z_d] (16x128), Dw1 [128,64] staged in LDS via TDM.
// K-loop split at 64 so each half reads one gathered row. Epilogue: relu + dot(Dw2) via
// 16-lane butterfly (bit4 untouched -> stays within the C-layout half).
__global__ void decoder_wmma(const float* __restrict__ z,
                             const int* __restrict__ eli,   // [2*P]
                             const float* __restrict__ Dw1, // [128,64]
                             const float* __restrict__ Db1, // [64]
                             const float* __restrict__ Dw2, // [64]
                             const float* __restrict__ Db2, // [1]
                             float* __restrict__ out, int P) {
  extern __shared__ float smem[];           // [128*64] Dw1
  const int wave = threadIdx.x >> 5;
  const int lane = threadIdx.x & 31;
  const int half = lane >> 4;
  const int l16  = lane & 15;

  if (wave == 0) {
    tdm_load_mat_to_lds(Dw1, 0u, 2 * HIDF);
    __builtin_amdgcn_s_wait_tensorcnt(0);
  }
  __syncthreads();

  const int pairBase = (blockIdx.x * (blockDim.x >> 5) + wave) * 16;
  if (pairBase >= P) return;

  const int p = pairBase + l16;
  const float* rs = z + (size_t)eli[p] * HIDF;
  const float* rd = z + (size_t)eli[P + p] * HIDF;
  const float* __restrict__ W_s = smem;

  v8f acc[4] = {};
  for (int k0 = 0; k0 < 64; k0 += 4) {
    const int ka = k0 + half * 2;
    v2f a;
    a.x = rs[ka];
    a.y = rs[ka + 1];
#pragma unroll
    for (int t = 0; t < 4; ++t) {
      const int col = t * 16 + l16;
      v2f b;
      b.x = W_s[ka * HIDF + col];
      b.y = W_s[(ka + 1) * HIDF + col];
      acc[t] = __builtin_amdgcn_wmma_f32_16x16x4_f32(false, a, false, b, (short)0,
                                                     acc[t], false, false);
    }
  }
  for (int k0 = 64; k0 < 128; k0 += 4) {
    const int ka = k0 + half * 2;
    v2f a;
    a.x = rd[ka - 64];
    a.y = rd[ka - 63];
#pragma unroll
    for (int t = 0; t < 4; ++t) {
      const int col = t * 16 + l16;
      v2f b;
      b.x = W_s[ka * HIDF + col];
      b.y = W_s[(ka + 1) * HIDF + col];
      acc[t] = __builtin_amdgcn_wmma_f32_16x16x4_f32(false, a, false, b, (short)0,
                                                     acc[t], false, false);
    }
  }

  const float db2 = Db2[0];
#pragma unroll
  for (int r = 0; r < 8; ++r) {
    float v = 0.0f;
#pragma unroll
    for (int t = 0; t < 4; ++t) {
      const int col = t * 16 + l16;
      float h = fmaxf(acc[t][r] + Db1[col], 0.0f);
      v += h * Dw2[col];
    }
    v += __shfl_xor(v, 1, 32);
    v += __shfl_xor(v, 2, 32);
    v += __shfl_xor(v, 4, 32);
    v += __shfl_xor(v, 8, 32);
    if (l16 == 0) out[pairBase + half * 8 + r] = v + db2;
  }
}

static inline int cdiv_l(long a, long b) { return (int)((a + b - 1) / b); }

extern "C" void kernel_launch(void* const* d_in, const int* in_sizes, int n_in,
                              void* d_out, int out_size, void* d_ws, size_t ws_size,
                              hipStream_t stream) {
  const float* x   = (const float*)d_in[0];
  const int*   ei  = (const int*)d_in[1];   // [2,E] flat
  const int*   eli = (const int*)d_in[2];   // [2,P] flat
  const float* Wl1 = (const float*)d_in[3];
  const float* Wr1 = (const float*)d_in[4];
  const float* b1  = (const float*)d_in[5];
  const float* Wl2 = (const float*)d_in[6];
  const float* Wr2 = (const float*)d_in[7];
  const float* b2  = (const float*)d_in[8];
  const float* Dw1 = (const float*)d_in[9];
  const float* Db1 = (const float*)d_in[10];
  const float* Dw2 = (const float*)d_in[11];
  const float* Db2 = (const float*)d_in[12];
  float* out = (float*)d_out;

  const int IN_F = 128;
  const int N = in_sizes[0] / IN_F;
  const int E = in_sizes[1] / 2;
  const int P = in_sizes[2] / 2;
  const int* src = ei;
  const int* dst = ei + E;

  const size_t nh = (size_t)N * HIDF;
  float* ws  = (float*)d_ws;
  float* hA  = ws;            // neighbor-transformed features (h @ Wl)
  float* hB  = ws + nh;       // self part (+bias); becomes h after finalize (in-place)
  float* agg = ws + 2 * nh;   // scatter accumulator
  float* hC  = ws + 3 * nh;   // layer-2 self part; becomes z (in-place)
  float* deg = ws + 4 * nh;   // [N]

  const int TPB = 256;
  const int wavesPerBlk = TPB / 32;
  const int gemmBlksN = cdiv_l((long)N / 16, wavesPerBlk);
  const int decBlks   = cdiv_l((long)P / 16, wavesPerBlk);

  const size_t lds1 = (size_t)2 * IN_F * HIDF * sizeof(float);  // 64 KB (Wa+Wb, K=128)
  const size_t lds2 = (size_t)2 * HIDF * HIDF * sizeof(float);  // 32 KB (K=64)
  const size_t ldsD = (size_t)2 * HIDF * HIDF * sizeof(float);  // 32 KB (Dw1)

  // degrees + zeroed accumulator (re-init every call; harness poisons ws)
  zero_f32<<<cdiv_l((long)N, TPB), TPB, 0, stream>>>(deg, N);
  zero_f32<<<cdiv_l((long)nh, TPB), TPB, 0, stream>>>(agg, (long)nh);
  degree_kernel<<<cdiv_l(E, TPB), TPB, 0, stream>>>(dst, deg, E);

  // Layer 1: transform BEFORE aggregating (mean-agg is linear) -> 64-dim gathers
  gemm_dual_wmma<<<gemmBlksN, TPB, lds1, stream>>>(x, Wl1, Wr1, b1, hA, hB, N, IN_F);
  scatter_add64<<<cdiv_l((long)E * 16, TPB), TPB, 0, stream>>>(hA, src, dst, agg, E);
  finalize_kernel<true><<<cdiv_l((long)nh, TPB), TPB, 0, stream>>>(agg, hB, deg, hB, (long)nh);

  // Layer 2
  zero_f32<<<cdiv_l((long)nh, TPB), TPB, 0, stream>>>(agg, (long)nh);
  gemm_dual_wmma<<<gemmBlksN, TPB, lds2, stream>>>(hB, Wl2, Wr2, b2, hA, hC, N, HIDF);
  scatter_add64<<<cdiv_l((long)E * 16, TPB), TPB, 0, stream>>>(hA, src, dst, agg, E);
  finalize_kernel<false><<<cdiv_l((long)nh, TPB), TPB, 0, stream>>>(agg, hC, deg, hC, (long)nh);

  // Decoder
  decoder_wmma<<<decBlks, TPB, ldsD, stream>>>(hC, eli, Dw1, Db1, Dw2, Db2, out, P);
}